// _InsertChannels_24111946399874
// MI455X (gfx1250) — compile-verified
//
#include <hip/hip_runtime.h>
#include <stdint.h>

// out = zeros(128,768,768); out[:, 256:, 256:] = rho   (idx[x] == x + 256)
// Pure bandwidth problem: 134 MB read + 302 MB write ~= 18.7 us at 23.3 TB/s.
// Interior copy uses gfx1250 async global<->LDS b128 ops (ASYNCcnt path).

typedef float __attribute__((ext_vector_type(4))) f32x4;

// Geometry in vec4 (16B) units:
//  in : [128][512][128] vec4
//  out: [128][768][192] vec4, interior at rows 256..767, vec-cols 64..191
#define IN_ROW_V    128          // 512 floats / 4
#define OUT_ROW_V   192          // 768 floats / 4
#define IN_IMG_V    (512 * IN_ROW_V)   // 65536 vec4 per batch image
#define BORDER_TOP  (256 * OUT_ROW_V)  // rows 0..255, full width  = 49152
#define BORDER_LEFT (512 * 64)         // rows 256..767, cols 0..255 = 32768
#define BORDER_PB   (BORDER_TOP + BORDER_LEFT)  // 81920 vec4 per batch

__global__ void ic_zero_border(f32x4* __restrict__ out4) {
    int t = blockIdx.x * blockDim.x + threadIdx.x;   // < 128*81920 = 10,485,760
    int b = t / BORDER_PB;
    int u = t - b * BORDER_PB;
    int r, cv;
    if (u < BORDER_TOP) {
        r  = u / OUT_ROW_V;
        cv = u - r * OUT_ROW_V;
    } else {
        int u2 = u - BORDER_TOP;
        r  = 256 + (u2 >> 6);
        cv = u2 & 63;
    }
    f32x4 z = {0.0f, 0.0f, 0.0f, 0.0f};
    // streaming store: no reuse, don't pollute L2
    __builtin_nontemporal_store(z, &out4[(b * 768 + r) * OUT_ROW_V + cv]);
}

// Each block moves 1024 consecutive interior vec4 (= 8 full input rows, 16 KB)
// through LDS with async b128 transfers. Thread k-th chunk is strided by 256
// so every issued instruction is a fully-coalesced 512 B wave access.
__global__ void ic_copy_interior_async(const f32x4* __restrict__ in4,
                                       f32x4* __restrict__ out4) {
    __shared__ f32x4 buf[1024];
    const int t    = threadIdx.x;
    const int base = blockIdx.x * 1024 + t;

#pragma unroll
    for (int k = 0; k < 4; ++k) {
        int v   = base + k * 256;
        int b   = v >> 16;            // / IN_IMG_V
        int rem = v & 0xFFFF;
        int r   = rem >> 7;           // / IN_ROW_V
        int c   = rem & 127;
        const f32x4* src = &in4[(b << 9 | r) * IN_ROW_V + c];  // (b*512+r)*128+c
        uint32_t lds = (uint32_t)(uintptr_t)&buf[t + k * 256]; // low 32b of flat LDS ptr = LDS byte offset
        asm volatile("global_load_async_to_lds_b128 %0, %1, off"
                     :: "v"(lds), "v"(src) : "memory");
    }

    // async loads must land in LDS before async stores read it
    asm volatile("s_wait_asynccnt 0" ::: "memory");

#pragma unroll
    for (int k = 0; k < 4; ++k) {
        int v   = base + k * 256;
        int b   = v >> 16;
        int rem = v & 0xFFFF;
        int r   = rem >> 7;
        int c   = rem & 127;
        f32x4* dst = &out4[(b * 768 + 256 + r) * OUT_ROW_V + 64 + c];
        uint32_t lds = (uint32_t)(uintptr_t)&buf[t + k * 256];
        asm volatile("global_store_async_from_lds_b128 %0, %1, off"
                     :: "v"(dst), "v"(lds) : "memory");
    }
    // s_endpgm performs an implicit wait-idle, draining outstanding async stores
}

extern "C" void kernel_launch(void* const* d_in, const int* in_sizes, int n_in,
                              void* d_out, int out_size, void* d_ws, size_t ws_size,
                              hipStream_t stream) {
    const f32x4* in4  = (const f32x4*)d_in[0];
    f32x4*       out4 = (f32x4*)d_out;

    // borders: 128 * 81920 vec4, one per thread
    int border_vec = 128 * BORDER_PB;                  // 10,485,760
    ic_zero_border<<<border_vec / 256, 256, 0, stream>>>(out4);

    // interior: 128 * 65536 vec4, 1024 per block
    int interior_vec = 128 * IN_IMG_V;                 // 8,388,608
    ic_copy_interior_async<<<interior_vec / 1024, 256, 0, stream>>>(in4, out4);
}